// MagnitudeAwareLinearAttention_3015067042126
// MI455X (gfx1250) — compile-verified
//
#include <hip/hip_runtime.h>
#include <hip/hip_bf16.h>
#include <math.h>
#include <stdint.h>

typedef __attribute__((ext_vector_type(2))) float v2f;
typedef __attribute__((ext_vector_type(8))) float v8f;

#define HEADS 8
#define HD 32
#define NPIX 4096

// LDS tile geometry for the staged GEMM
#define A_STRIDE 36   // 64 rows x 32 k, padded: 16B-aligned rows, conflict-free
#define B_STRIDE 34   // 64 cols x 32 k (transposed), 8B-aligned, conflict-free
#define A_FLOATS (64 * A_STRIDE)            // 2304
#define BUF_FLOATS (A_FLOATS + 64 * B_STRIDE)  // 2304 + 2176 = 4480

// ---------------------------------------------------------------------------
// Async-stage one K-chunk (32) of A (64 x 32, row-major) and B (32 x 64,
// stored transposed [n][k]) into LDS via global_load_async_to_lds (ASYNCcnt).
// ---------------------------------------------------------------------------
__device__ __forceinline__ void gemm_stage_async(
    float* __restrict__ sbuf, const float* __restrict__ gA,
    const float* __restrict__ gB, int Kdim, int N, int t) {
  // A: 512 float4s, global rows contiguous in k
#pragma unroll
  for (int j = 0; j < 2; ++j) {
    const int i = t + j * 256;
    const int row = i >> 3, c4 = (i & 7) * 4;
    const unsigned la = (unsigned)(size_t)&sbuf[row * A_STRIDE + c4];
    const uint64_t ga = (uint64_t)(gA + (size_t)row * Kdim + c4);
    asm volatile("global_load_async_to_lds_b128 %0, %1, off"
                 :: "v"(la), "v"(ga) : "memory");
  }
  // B: 2048 scalars, coalesced along n in global, scattered transposed in LDS
#pragma unroll
  for (int j = 0; j < 8; ++j) {
    const int i = t + j * 256;
    const int kr = i >> 6, n = i & 63;
    const unsigned lb = (unsigned)(size_t)&sbuf[A_FLOATS + n * B_STRIDE + kr];
    const uint64_t gb = (uint64_t)(gB + (size_t)kr * N + n);
    asm volatile("global_load_async_to_lds_b32 %0, %1, off"
                 :: "v"(lb), "v"(gb) : "memory");
  }
}

// ---------------------------------------------------------------------------
// Channel-mixing GEMM:  Out[b,o,n] = sum_c W[o,c] * X[b,c,n] + bias[o]
// 64x64 block tile, 8 waves (4 M-bands x 2 N-halves), K chunked by 32 with
// double-buffered async-to-LDS staging; V_WMMA_F32_16X16X4_F32 from LDS.
// grid: x = N/64, y = Cout/64, z = batch; block = 256
// ---------------------------------------------------------------------------
__global__ __launch_bounds__(256) void gemm_wmma_lds(
    const float* __restrict__ W, const float* __restrict__ X,
    const float* __restrict__ bias, float* __restrict__ Out,
    int Cout, int K, int N) {
  __shared__ float smem[2][BUF_FLOATS];
  const int t = threadIdx.x;
  const int lane = t & 31, wv = t >> 5;
  const int mt = wv & 3;        // 16-row band within 64
  const int nb = wv >> 2;       // 32-col half within 64
  const int n0 = blockIdx.x * 64, o0 = blockIdx.y * 64, b = blockIdx.z;
  const float* __restrict__ Xb = X + (size_t)b * K * N;

  const int m = lane & 15;         // A: M row, B/D: N column
  const int kk = (lane >> 4) * 2;  // K pair: lanes 0-15 -> {0,1}, 16-31 -> {2,3}

  v8f c0 = {}, c1 = {};
  const int NC = K / 32;

  gemm_stage_async(smem[0], W + (size_t)o0 * K, Xb + n0, K, N, t);
  for (int ch = 0; ch < NC; ++ch) {
    const int cur = ch & 1;
    asm volatile("s_wait_asynccnt 0x0" ::: "memory");
    __syncthreads();                       // chunk `ch` resident in smem[cur]
    if (ch + 1 < NC)
      gemm_stage_async(smem[cur ^ 1], W + (size_t)o0 * K + (ch + 1) * 32,
                       Xb + (size_t)(ch + 1) * 32 * N + n0, K, N, t);

    const float* __restrict__ sA = smem[cur];
    const float* __restrict__ sB = smem[cur] + A_FLOATS;
#pragma unroll
    for (int k4 = 0; k4 < 32; k4 += 4) {
      v2f a  = *(const v2f*)(sA + (mt * 16 + m) * A_STRIDE + k4 + kk);
      v2f b0 = *(const v2f*)(sB + (nb * 32 + m) * B_STRIDE + k4 + kk);
      v2f b1 = *(const v2f*)(sB + (nb * 32 + 16 + m) * B_STRIDE + k4 + kk);
      c0 = __builtin_amdgcn_wmma_f32_16x16x4_f32(false, a, false, b0, (short)0,
                                                 c0, false, false);
      c1 = __builtin_amdgcn_wmma_f32_16x16x4_f32(false, a, false, b1, (short)0,
                                                 c1, false, false);
    }
    __syncthreads();                       // done reading smem[cur]
  }

  // C/D layout: vgpr r, lane L -> M = r + 8*(L>=16), N = L%16
  const int mb = (lane >> 4) * 8;
  const int nc0 = n0 + nb * 32 + m;
#pragma unroll
  for (int r = 0; r < 8; ++r) {
    const int oc = o0 + mt * 16 + r + mb;
    const float bs = bias[oc];
    Out[((size_t)b * Cout + oc) * N + nc0] = c0[r] + bs;
    Out[((size_t)b * Cout + oc) * N + nc0 + 16] = c1[r] + bs;
  }
}

// ---------------------------------------------------------------------------
// Repack qkv channels (b, 3C, n) -> per-head row-major (b, h, n, d) buffers.
// q,k get elu(x)+1 here (pre-RoPE values, as the reference's z requires).
// ---------------------------------------------------------------------------
__global__ __launch_bounds__(256) void prep_qkv(
    const float* __restrict__ qkvo, float* __restrict__ qb,
    float* __restrict__ kb, float* __restrict__ vb) {
  const size_t idx = (size_t)blockIdx.x * blockDim.x + threadIdx.x;
  const int n = (int)(idx & (NPIX - 1));
  const int c3 = (int)((idx >> 12) % 768);          // 0..767
  const int b = (int)(idx / ((size_t)NPIX * 768));
  const int grp = c3 >> 8;        // 0=q 1=k 2=v
  const int c = c3 & 255;
  const int h = c >> 5, d = c & 31;

  const float v = qkvo[((size_t)b * 1024 + c3) * NPIX + n];
  const size_t dst = (((size_t)(b * HEADS + h) * NPIX) + n) * HD + d;
  if (grp == 0) {
    qb[dst] = v > 0.0f ? v + 1.0f : expf(v);        // elu(v)+1
  } else if (grp == 1) {
    kb[dst] = v > 0.0f ? v + 1.0f : expf(v);
  } else {
    vb[dst] = v;
  }
}

// kmean[bh,d] = scale * mean_n k[bh,n,d]; vmean[bh,d] = mean_n v[bh,n,d]
__global__ __launch_bounds__(256) void means_kernel(
    const float* __restrict__ kb, const float* __restrict__ vb,
    float* __restrict__ kmean, float* __restrict__ vmean, float scale) {
  const int bh = blockIdx.x;
  const int t = threadIdx.x;
  const int d = t & 31, rg = t >> 5;
  const float* kp = kb + (size_t)bh * NPIX * HD;
  const float* vp = vb + (size_t)bh * NPIX * HD;
  float ka = 0.f, va = 0.f;
  for (int n = rg; n < NPIX; n += 8) {
    ka += kp[(size_t)n * HD + d];
    va += vp[(size_t)n * HD + d];
  }
  __shared__ float sk[256], sv[256];
  sk[t] = ka; sv[t] = va;
  __syncthreads();
  if (t < 32) {
    float a = 0.f, bsum = 0.f;
#pragma unroll
    for (int i = 0; i < 8; ++i) { a += sk[i * 32 + t]; bsum += sv[i * 32 + t]; }
    kmean[bh * 32 + t] = a * (scale / (float)NPIX);
    vmean[bh * 32 + t] = bsum / (float)NPIX;
  }
}

// z[bh,n] = dot(q_pre_rope[bh,n,:], kmean[bh,:])
__global__ __launch_bounds__(256) void z_kernel(
    const float* __restrict__ qb, const float* __restrict__ kmean,
    float* __restrict__ z) {
  const size_t idx = (size_t)blockIdx.x * blockDim.x + threadIdx.x;
  const int bh = (int)(idx >> 12);
  const float* q = qb + idx * HD;
  const float* km = kmean + bh * 32;
  float acc = 0.f;
#pragma unroll
  for (int d = 0; d < HD; ++d) acc += q[d] * km[d];
  z[idx] = acc;
}

// theta_shift in place; k additionally scaled by s2 = scale/N
__global__ __launch_bounds__(256) void rope_kernel(
    float* __restrict__ qb, float* __restrict__ kb,
    const float* __restrict__ sn, const float* __restrict__ cs, float s2) {
  const size_t idx = (size_t)blockIdx.x * blockDim.x + threadIdx.x;
  const int i = (int)(idx & 15);
  const int n = (int)((idx >> 4) & (NPIX - 1));
  const size_t bh = idx >> 16;
  const size_t base = (bh * NPIX + n) * HD + 2 * i;
  const float c0 = cs[n * HD + 2 * i], c1 = cs[n * HD + 2 * i + 1];
  const float s0 = sn[n * HD + 2 * i], s1 = sn[n * HD + 2 * i + 1];
  const float q0 = qb[base], q1 = qb[base + 1];
  qb[base]     = q0 * c0 - q1 * s0;
  qb[base + 1] = q1 * c1 + q0 * s1;
  const float k0 = kb[base], k1 = kb[base + 1];
  kb[base]     = (k0 * c0 - k1 * s0) * s2;
  kb[base + 1] = (k1 * c1 + k0 * s1) * s2;
}

// kv[bh,d,e] = sum_l k'[bh,l,d] * v[bh,l,e]   (32x32 per head-batch)
__global__ __launch_bounds__(256) void kv_kernel(
    const float* __restrict__ kb, const float* __restrict__ vb,
    float* __restrict__ kv) {
  const int bh = blockIdx.x;
  const int t = threadIdx.x;
  const int d = t & 31, e4 = (t >> 5) * 4;
  const float* kp = kb + (size_t)bh * NPIX * HD;
  const float* vp = vb + (size_t)bh * NPIX * HD;
  __shared__ float sk[32][33], sv[32][33];
  float a0 = 0.f, a1 = 0.f, a2 = 0.f, a3 = 0.f;
  for (int l0 = 0; l0 < NPIX; l0 += 32) {
    __syncthreads();
#pragma unroll
    for (int j = 0; j < 4; ++j) {
      const int off = t + j * 256;
      sk[off >> 5][off & 31] = kp[(size_t)(l0 + (off >> 5)) * HD + (off & 31)];
      sv[off >> 5][off & 31] = vp[(size_t)(l0 + (off >> 5)) * HD + (off & 31)];
    }
    __syncthreads();
#pragma unroll
    for (int ll = 0; ll < 32; ++ll) {
      const float kd = sk[ll][d];
      a0 += kd * sv[ll][e4 + 0];
      a1 += kd * sv[ll][e4 + 1];
      a2 += kd * sv[ll][e4 + 2];
      a3 += kd * sv[ll][e4 + 3];
    }
  }
  float* o = kv + (size_t)bh * 1024 + d * 32 + e4;
  o[0] = a0; o[1] = a1; o[2] = a2; o[3] = a3;
}

// res[b, h*32+e, l] = (q' @ kv)[l,e] * (1 + 1/(z+1e-6)) - z[l]*vmean[e]
__global__ __launch_bounds__(256) void attn_out_wmma(
    const float* __restrict__ qb, const float* __restrict__ kv,
    const float* __restrict__ z, const float* __restrict__ vmean,
    float* __restrict__ res) {
  const int lane = threadIdx.x;
  const int bh = blockIdx.y;
  const int l0 = (blockIdx.x * 8 + threadIdx.y) * 16;
  const int b = bh >> 3, h = bh & 7;

  const float* __restrict__ q = qb + (size_t)bh * NPIX * HD;
  const float* __restrict__ kvm = kv + (size_t)bh * 1024;  // [d][e] row-major

  const int m = lane & 15;
  const int kk = (lane >> 4) * 2;

  v8f c0 = {}, c1 = {};
#pragma unroll
  for (int k0 = 0; k0 < HD; k0 += 4) {
    v2f a = *(const v2f*)(q + (size_t)(l0 + m) * HD + k0 + kk);
    v2f b0, b1;
    b0.x = kvm[(k0 + kk) * 32 + m];
    b0.y = kvm[(k0 + kk + 1) * 32 + m];
    b1.x = kvm[(k0 + kk) * 32 + 16 + m];
    b1.y = kvm[(k0 + kk + 1) * 32 + 16 + m];
    c0 = __builtin_amdgcn_wmma_f32_16x16x4_f32(false, a, false, b0, (short)0,
                                               c0, false, false);
    c1 = __builtin_amdgcn_wmma_f32_16x16x4_f32(false, a, false, b1, (short)0,
                                               c1, false, false);
  }

  const float* zrow = z + (size_t)bh * NPIX + l0;
  const float vme0 = vmean[bh * 32 + m];
  const float vme1 = vmean[bh * 32 + 16 + m];
  const int mb = (lane >> 4) * 8;
#pragma unroll
  for (int r = 0; r < 8; ++r) {
    const int l = l0 + r + mb;
    const float zz = zrow[r + mb];
    const float sc = 1.0f + 1.0f / (zz + 1e-6f);
    res[((size_t)(b * 256 + h * 32 + m)) * NPIX + l] = c0[r] * sc - zz * vme0;
    res[((size_t)(b * 256 + h * 32 + 16 + m)) * NPIX + l] = c1[r] * sc - zz * vme1;
  }
}

// res[b,c,n] = (res[b,c,n] + lepe(v)[b,c,n]) * o[b,c,n]   (in place)
__global__ __launch_bounds__(256) void lepe_mul_kernel(
    const float* __restrict__ qkvo, const float* __restrict__ w_lepe,
    const float* __restrict__ b_lepe, float* __restrict__ res) {
  const int n = blockIdx.x * 256 + threadIdx.x;
  const int c = blockIdx.y, b = blockIdx.z;
  const int hh = n >> 6, ww = n & 63;
  const float* __restrict__ vch = qkvo + ((size_t)b * 1024 + 512 + c) * NPIX;
  const float* __restrict__ och = qkvo + ((size_t)b * 1024 + 768 + c) * NPIX;
  const float* __restrict__ wl = w_lepe + c * 25;
  float acc = b_lepe[c];
#pragma unroll
  for (int dy = 0; dy < 5; ++dy) {
    const int y = hh + dy - 2;
    if ((unsigned)y < 64u) {
#pragma unroll
      for (int dx = 0; dx < 5; ++dx) {
        const int x = ww + dx - 2;
        if ((unsigned)x < 64u) acc += wl[dy * 5 + dx] * vch[y * 64 + x];
      }
    }
  }
  const size_t oi = ((size_t)b * 256 + c) * NPIX + n;
  res[oi] = (res[oi] + acc) * och[n];
}

// ---------------------------------------------------------------------------
extern "C" void kernel_launch(void* const* d_in, const int* in_sizes, int n_in,
                              void* d_out, int out_size, void* d_ws,
                              size_t ws_size, hipStream_t stream) {
  const float* x      = (const float*)d_in[0];
  const float* sn     = (const float*)d_in[1];
  const float* cs     = (const float*)d_in[2];
  const float* w_qkvo = (const float*)d_in[3];
  const float* b_qkvo = (const float*)d_in[4];
  const float* w_lepe = (const float*)d_in[5];
  const float* b_lepe = (const float*)d_in[6];
  const float* w_proj = (const float*)d_in[7];
  const float* b_proj = (const float*)d_in[8];
  float* out = (float*)d_out;

  const int B = 8, C = 256, N = NPIX;
  const float scale = 0.17677669529663687f;  // 32^-0.5
  const float s2 = scale / (float)N;         // s*s, s = sqrt(scale/N)

  float* ws    = (float*)d_ws;
  float* qkvo  = ws;                    // B*1024*N      = 33,554,432
  float* qb    = qkvo  + 33554432;      // B*8*N*32      =  8,388,608
  float* kb    = qb    + 8388608;
  float* vb    = kb    + 8388608;
  float* res   = vb    + 8388608;       // B*C*N
  float* zb    = res   + 8388608;       // 64*N          =    262,144
  float* kmean = zb    + 262144;        // 64*32
  float* vmean = kmean + 2048;
  float* kvb   = vmean + 2048;          // 64*1024

  // 1) qkvo = W_qkvo @ x + b  (async-LDS staged f32 WMMA GEMM)
  gemm_wmma_lds<<<dim3(N / 64, 1024 / 64, B), 256, 0, stream>>>(
      w_qkvo, x, b_qkvo, qkvo, 1024, 256, N);
  // 2) repack + elu(q,k)+1
  prep_qkv<<<(B * 768 * N) / 256, 256, 0, stream>>>(qkvo, qb, kb, vb);
  // 3) means (scale folded into kmean)
  means_kernel<<<64, 256, 0, stream>>>(kb, vb, kmean, vmean, scale);
  // 4) z = q . kmean
  z_kernel<<<(64 * N) / 256, 256, 0, stream>>>(qb, kmean, zb);
  // 5) RoPE q,k (k *= s^2)
  rope_kernel<<<(64 * N * 16) / 256, 256, 0, stream>>>(qb, kb, sn, cs, s2);
  // 6) kv = k'^T v
  kv_kernel<<<64, 256, 0, stream>>>(kb, vb, kvb);
  // 7) res = (q' @ kv) * (1+1/(z+eps)) - z*vmean  (f32 WMMA)
  attn_out_wmma<<<dim3(N / 128, 64), dim3(32, 8), 0, stream>>>(qb, kvb, zb,
                                                               vmean, res);
  // 8) res = (res + lepe) * o
  lepe_mul_kernel<<<dim3(N / 256, C, B), 256, 0, stream>>>(qkvo, w_lepe,
                                                           b_lepe, res);
  // 9) out = W_proj @ res + b_proj (async-LDS staged f32 WMMA GEMM)
  gemm_wmma_lds<<<dim3(N / 64, C / 64, B), 256, 0, stream>>>(
      w_proj, res, b_proj, out, 256, 256, N);
}